// APEV_25701084299541
// MI455X (gfx1250) — compile-verified
//
#include <hip/hip_runtime.h>
#include <math.h>

#define RC_F     5.2f
#define N_BATCH  32
#define N_ATOMS  256
#define N_ETA    4
#define N_SHF    8
#define N_OUT    32     // N_ETA * N_SHF
#define EPB      256    // edges handled per block (== blockDim.x)
#define ROWP     36     // padded LDS row stride (floats) to break bank conflicts

// grid = (E/EPB, B); one block = 256 threads = 8 wave32 = 256 edges of batch blockIdx.y.
// Batch coords (256 atoms x 3 f32) are async-copied into LDS (float4-padded per atom
// -> single aligned ds_load_b128 per gather). Output tile is staged in LDS and
// written back fully coalesced (each wave store covers one contiguous 512B span).
__global__ __launch_bounds__(EPB) void aev_radial_kernel(
    const int*   __restrict__ conn,     // [B*E*2]
    const float* __restrict__ coords,   // [B*A*3]
    const float* __restrict__ etaR,     // [4]
    const float* __restrict__ shfR,     // [8]
    int*         __restrict__ out_conn, // [B*E*2]  (passthrough)
    float*       __restrict__ out_y)    // [B*E*32]
{
    __shared__ __align__(16) float s_coords[N_ATOMS * 4];   // 4KB
    __shared__ __align__(16) float s_out[EPB * ROWP];       // 36KB staging tile

    const int t = threadIdx.x;
    const int b = blockIdx.y;                                // batch (no division)
    const int E = gridDim.x * EPB;                           // edges per batch

    // ---- stage batch b's coords into LDS via CDNA5 async copy ------------
    // one atom per thread: global t*12+{0,4,8}  ->  LDS t*16+{0,4,8}
    const float* gbase = coords + (size_t)b * (N_ATOMS * 3);
    {
        unsigned goff = (unsigned)t * 12u;
        __attribute__((address_space(3))) float* lp =
            (__attribute__((address_space(3))) float*)&s_coords[t * 4];
        asm volatile("global_load_async_to_lds_b32 %0, %1, %2"
                     :: "v"(lp),     "v"(goff),      "s"(gbase) : "memory");
        asm volatile("global_load_async_to_lds_b32 %0, %1, %2"
                     :: "v"(lp + 1), "v"(goff + 4u), "s"(gbase) : "memory");
        asm volatile("global_load_async_to_lds_b32 %0, %1, %2"
                     :: "v"(lp + 2), "v"(goff + 8u), "s"(gbase) : "memory");
    }
    asm volatile("s_wait_asynccnt 0" ::: "memory");   // drain this wave's ASYNCcnt
    __syncthreads();                                   // LDS visible workgroup-wide

    // ---- uniform small tables (scalarizable) ------------------------------
    float eta[N_ETA], shf[N_SHF];
    #pragma unroll
    for (int i = 0; i < N_ETA; ++i) eta[i] = etaR[i];
    #pragma unroll
    for (int j = 0; j < N_SHF; ++j) shf[j] = shfR[j];

    // ---- per-edge work ----------------------------------------------------
    const size_t e = (size_t)b * E + (size_t)blockIdx.x * EPB + t;
    int2 cp = ((const int2*)conn)[e];
    ((int2*)out_conn)[e] = cp;                // connectivity passthrough

    const bool ok = (cp.x != -1);
    const int ia = ok ? cp.x : 0;
    const int id = ok ? cp.y : 0;

    float4 ca = *(const float4*)&s_coords[ia * 4];   // ds_load_b128
    float4 cd = *(const float4*)&s_coords[id * 4];   // ds_load_b128
    float dx = ca.x - cd.x, dy = ca.y - cd.y, dz = ca.z - cd.z;
    float dist = sqrtf(dx * dx + dy * dy + dz * dz);

    float fc  = (dist <= RC_F)
              ? (0.5f * __cosf(3.14159265358979323846f * dist * (1.0f / RC_F)) + 0.5f)
              : 0.0f;
    float pre = ok ? (0.25f * fc) : 0.0f;

    float out[N_OUT];
    #pragma unroll
    for (int j = 0; j < N_SHF; ++j) {
        float td = dist - shf[j];
        float x  = td * td;
        #pragma unroll
        for (int i = 0; i < N_ETA; ++i)
            out[i * N_SHF + j] = pre * __expf(-eta[i] * x);   // v_exp_f32
    }

    // ---- stage row in LDS (padded stride), then coalesced write-back ------
    {
        float4* sp = (float4*)&s_out[t * ROWP];
        #pragma unroll
        for (int q = 0; q < 8; ++q) sp[q] = ((const float4*)out)[q];
    }
    __syncthreads();

    float4* ytile = (float4*)(out_y +
        ((size_t)b * E + (size_t)blockIdx.x * EPB) * (size_t)N_OUT);
    #pragma unroll
    for (int k = 0; k < 8; ++k) {
        int g    = t + k * EPB;        // float4 id in tile: 0..2047
        int edge = g >> 3;
        int q    = g & 7;
        ytile[g] = *(const float4*)&s_out[edge * ROWP + q * 4];  // coalesced 512B/wave
    }
}

extern "C" void kernel_launch(void* const* d_in, const int* in_sizes, int n_in,
                              void* d_out, int out_size, void* d_ws, size_t ws_size,
                              hipStream_t stream) {
    const int*   conn   = (const int*)  d_in[0];   // [B,E,2] int32
    const float* coords = (const float*)d_in[1];   // [B,A,3] f32
    const float* etaR   = (const float*)d_in[2];   // [4]
    const float* shfR   = (const float*)d_in[3];   // [8]

    const int n_edges_total   = in_sizes[0] / 2;            // B*E
    const int edges_per_batch = n_edges_total / N_BATCH;    // E (reference fixes B=32)

    int*   out_conn = (int*)d_out;                               // tuple elem 0
    float* out_y    = (float*)d_out + (size_t)n_edges_total * 2; // tuple elem 1

    dim3 grid(edges_per_batch / EPB, N_BATCH);   // (128, 32)
    aev_radial_kernel<<<grid, EPB, 0, stream>>>(
        conn, coords, etaR, shfR, out_conn, out_y);
}